// SparsemaxBisectLoss_83923660964515
// MI455X (gfx1250) — compile-verified
//
#include <hip/hip_runtime.h>

// Problem constants (match reference shapes).
#define NCOL         32000
#define THREADS      320            // 10 wave32 waves per block
#define WAVES        (THREADS / 32)
#define RSLOTS       (2 * WAVES)    // two half-wave partials per wave (20)
#define V4PT         25             // 25 float4 = 100 floats/thread; 320*100 = 32000
#define N_ITER       50
#define IGNORE_INDEX (-100)

typedef __attribute__((ext_vector_type(2))) float v2f;
typedef __attribute__((ext_vector_type(8))) float v8f;

// Partially sum the 32 lane-partials of a wave with one V_WMMA_F32_16X16X4_F32.
// A: VGPR0 = partial, VGPR1 = 0 -> each lane's partial occupies exactly one
// (m,k) slot of the 16x4 A matrix. B = all ones -> D[m][n] = rowsum(m).
// Lane n (0..15) holds D[0..7][n], lane n+16 holds D[8..15][n]; summing the
// 8 dest VGPRs leaves the wave total split across the xor-16 lane pair:
//   lanes 0..15 : sum of partials from lanes 0..7  and 16..23
//   lanes 16..31: sum of partials from lanes 8..15 and 24..31
// Lanes 0 and 16 publish both halves to LDS (no ds_bpermute round-trip).
__device__ __forceinline__ float wave_halfsum_wmma(float partial) {
    v2f a; a.x = partial; a.y = 0.0f;
    v2f b; b.x = 1.0f;    b.y = 1.0f;
    v8f c = {};
    c = __builtin_amdgcn_wmma_f32_16x16x4_f32(
        /*neg_a=*/false, a, /*neg_b=*/false, b,
        /*c_mod=*/(short)0, c, /*reuse_a=*/false, /*reuse_b=*/false);
    return ((c[0] + c[1]) + (c[2] + c[3])) + ((c[4] + c[5]) + (c[6] + c[7]));
}

// Block-wide sum with ONE barrier per call via phase double-buffering:
// phase p writes slot set (p&1); the p+1 write goes to the other slot set,
// and the p+2 reuse of this slot set is ordered by the phase p+1 barrier.
// Broadcast combine: 5 x ds_load_b128 + pairwise tree (depth ~5) instead of
// 20 scalar loads + a 20-deep serial add chain on the inter-iteration
// critical path. Every thread returns the identical total (uniform control
// flow; EXEC stays all-ones for the WMMA).
__device__ __forceinline__ float block_sum_phased(float red[2][RSLOTS],
                                                  float v, int phase) {
    const int tid = threadIdx.x;
    const float s8 = wave_halfsum_wmma(v);
    if ((tid & 15) == 0)                       // lanes 0 and 16 of each wave
        red[phase & 1][tid >> 4] = s8;         // slot = 2*wave + laneHalf
    __syncthreads();
    const float4* r4 = (const float4*)red[phase & 1];
    const float4 t0 = r4[0];
    const float4 t1 = r4[1];
    const float4 t2 = r4[2];
    const float4 t3 = r4[3];
    const float4 t4 = r4[4];
    const float a0 = (t0.x + t0.y) + (t0.z + t0.w);
    const float a1 = (t1.x + t1.y) + (t1.z + t1.w);
    const float a2 = (t2.x + t2.y) + (t2.z + t2.w);
    const float a3 = (t3.x + t3.y) + (t3.z + t3.w);
    const float a4 = (t4.x + t4.y) + (t4.z + t4.w);
    return ((a0 + a1) + (a2 + a3)) + a4;
}

__device__ __forceinline__ float block_max(float v, float* mred) {
    const int tid = threadIdx.x;
#pragma unroll
    for (int off = 16; off > 0; off >>= 1)
        v = fmaxf(v, __shfl_xor(v, off, 32));
    if ((tid & 31) == 0) mred[tid >> 5] = v;
    __syncthreads();
    float m = mred[0];
#pragma unroll
    for (int i = 1; i < WAVES; ++i) m = fmaxf(m, mred[i]);
    __syncthreads();                           // mred dead after this
    return m;
}

__global__ __launch_bounds__(THREADS)
void sparsemax_loss_rows(const float* __restrict__ X,
                         const int* __restrict__ target,
                         float* __restrict__ row_loss,
                         float* __restrict__ row_valid) {
    __shared__ __align__(16) float red[2][RSLOTS];
    __shared__ float mred[WAVES];
    const int row = blockIdx.x;
    const int tid = threadIdx.x;
    const float* __restrict__ xrow  = X + (size_t)row * NCOL;
    const float4* __restrict__ xrow4 = (const float4*)xrow;

    // Single coalesced HBM pass (25 x global_load_b128 per lane); the row
    // becomes register-resident (100 VGPRs/lane) and is never re-read.
    float4 v[V4PT];
#pragma unroll
    for (int k = 0; k < V4PT; ++k)
        v[k] = xrow4[tid + k * THREADS];

    // Row max.
    float m = -3.402823466e38f;
#pragma unroll
    for (int k = 0; k < V4PT; ++k)
        m = fmaxf(m, fmaxf(fmaxf(v[k].x, v[k].y), fmaxf(v[k].z, v[k].w)));
    m = block_max(m, mred);

    // Bisection (bit-faithful to reference: tau_lo = max-1, tau_hi = max-1/d,
    // dm = tau_hi - tau_lo, 50 halvings, f >= 0 keeps the midpoint).
    float tau_lo = m - 1.0f;
    const float tau_hi = m - (float)(1.0 / (double)NCOL);
    float dm    = tau_hi - tau_lo;
    float tau_m = tau_lo;
    for (int it = 0; it < N_ITER; ++it) {
        dm *= 0.5f;
        tau_m = tau_lo + dm;
        // Two accumulators: chain length 50 < the 150-slot dual-issue bound,
        // so the dependence chain stays off the critical path without SLP
        // packing / register blow-up past 256 VGPRs.
        float pa = 0.0f, pb = 0.0f;
#pragma unroll
        for (int k = 0; k < V4PT; ++k) {
            pa += fmaxf(v[k].x - tau_m, 0.0f);
            pb += fmaxf(v[k].y - tau_m, 0.0f);
            pa += fmaxf(v[k].z - tau_m, 0.0f);
            pb += fmaxf(v[k].w - tau_m, 0.0f);
        }
        const float f = block_sum_phased(red, pa + pb, it) - 1.0f;
        tau_lo = (f >= 0.0f) ? tau_m : tau_lo;   // uniform update
    }

    // Final sweep: S = sum p, S2 = sum p^2, SX = sum p*x (p un-normalized).
    float s1 = 0.0f, s2 = 0.0f, sx = 0.0f;
#pragma unroll
    for (int k = 0; k < V4PT; ++k) {
        float p;
        p = fmaxf(v[k].x - tau_m, 0.0f); s1 += p; s2 += p * p; sx += p * v[k].x;
        p = fmaxf(v[k].y - tau_m, 0.0f); s1 += p; s2 += p * p; sx += p * v[k].y;
        p = fmaxf(v[k].z - tau_m, 0.0f); s1 += p; s2 += p * p; sx += p * v[k].z;
        p = fmaxf(v[k].w - tau_m, 0.0f); s1 += p; s2 += p * p; sx += p * v[k].w;
    }
    const float S1 = block_sum_phased(red, s1, N_ITER + 0);
    const float S2 = block_sum_phased(red, s2, N_ITER + 1);
    const float SX = block_sum_phased(red, sx, N_ITER + 2);

    if (tid == 0) {
        const int  t     = target[row];
        const bool valid = (t != IGNORE_INDEX);
        const int  tt    = valid ? t : 0;
        const float xt   = xrow[tt];
        const float inv  = 1.0f / S1;                    // ensure_sum_one
        const float omega = 0.5f * (1.0f - S2 * inv * inv);
        const float loss  = omega + SX * inv - xt;       // + dot(p - onehot, x)
        row_loss[row]  = valid ? loss : 0.0f;
        row_valid[row] = valid ? 1.0f : 0.0f;
    }
}

__global__ __launch_bounds__(256)
void sparsemax_final_reduce(const float* __restrict__ row_loss,
                            const float* __restrict__ row_valid,
                            float* __restrict__ out, int nrows) {
    __shared__ float rs[8], rc[8];
    const int tid = threadIdx.x;
    float s = 0.0f, c = 0.0f;
    for (int i = tid; i < nrows; i += 256) {
        s += row_loss[i];
        c += row_valid[i];
    }
#pragma unroll
    for (int off = 16; off > 0; off >>= 1) {
        s += __shfl_xor(s, off, 32);
        c += __shfl_xor(c, off, 32);
    }
    if ((tid & 31) == 0) { rs[tid >> 5] = s; rc[tid >> 5] = c; }
    __syncthreads();
    if (tid == 0) {
        float S = 0.0f, C = 0.0f;
#pragma unroll
        for (int i = 0; i < 8; ++i) { S += rs[i]; C += rc[i]; }
        out[0] = S / fmaxf(C, 1.0f);   // mean over valid samples
    }
}

extern "C" void kernel_launch(void* const* d_in, const int* in_sizes, int n_in,
                              void* d_out, int out_size, void* d_ws, size_t ws_size,
                              hipStream_t stream) {
    const float* X      = (const float*)d_in[0];
    const int*   target = (const int*)d_in[1];
    const int    nrows  = in_sizes[1];          // 4096
    float* row_loss  = (float*)d_ws;            // nrows floats
    float* row_valid = row_loss + nrows;        // nrows floats

    sparsemax_loss_rows<<<nrows, THREADS, 0, stream>>>(X, target, row_loss, row_valid);
    sparsemax_final_reduce<<<1, 256, 0, stream>>>(row_loss, row_valid, (float*)d_out, nrows);
}